// Cycle_path_50869592655534
// MI455X (gfx1250) — compile-verified
//
#include <hip/hip_runtime.h>

#define HID 128
#define NN 50000
#define NCYC 10000
#define NPATH 15000
#define TILE_M 48
#define LDX 260   // 256 + 4 padding floats: stride%64banks = 4 -> conflict-free A reads

typedef __attribute__((ext_vector_type(2))) float v2f;
typedef __attribute__((ext_vector_type(8))) float v8f;
typedef int v4i __attribute__((vector_size(4 * sizeof(int))));

#if __has_builtin(__builtin_amdgcn_global_load_async_to_lds_b128)
#define HAVE_ASYNC_LDS 1
#else
#define HAVE_ASYNC_LDS 0
#endif

// ---------------- scatter: rep rows -> node feature table ----------------
__global__ void scatter_add_kernel(const float* __restrict__ rep,
                                   const int* __restrict__ nodes,
                                   float* __restrict__ nodefeat,
                                   int nrows) {
    int row = blockIdx.x;
    if (row >= nrows) return;
    int node = nodes[row];
    int c = threadIdx.x;  // 128 threads
    atomicAdd(&nodefeat[(size_t)node * HID + c], rep[(size_t)row * HID + c]);
}

// ---------------- per-subgraph sum of gathered node features ----------------
__global__ void subgraph_sum_kernel(const float* __restrict__ nodefeat,
                                    const int* __restrict__ nodes,
                                    float* __restrict__ sums,
                                    int nsub, int size) {
    int sub = blockIdx.x;
    if (sub >= nsub) return;
    int c = threadIdx.x;  // 128 threads
    float acc = 0.f;
    for (int k = 0; k < size; ++k) {
        int node = nodes[sub * size + k];
        acc += nodefeat[(size_t)node * HID + c];
    }
    sums[(size_t)sub * HID + c] = acc;
}

// ---------------- build effective weight, packed as (kpair, col, 2) ----------------
// Waug[k][col], k<128: w_loc[k][col]
//              k>=128: w_loc[k][col] + w_glb[k-128][col] + size*w_glb[k][col]
// packed so that the per-lane B fragment (elements (k,col),(k+1,col)) is one float2
__global__ void pack_weights_kernel(const float* __restrict__ w_loc,
                                    const float* __restrict__ w_glb,
                                    float* __restrict__ wp,
                                    float scale) {
    int idx = blockIdx.x * blockDim.x + threadIdx.x;  // element (k,col)
    if (idx >= 2 * HID * HID) return;
    int k   = idx / HID;
    int col = idx % HID;
    float v = w_loc[idx];
    if (k >= HID) v += w_glb[idx - HID * HID] + scale * w_glb[idx];
    wp[((size_t)(k >> 1) * HID + col) * 2 + (k & 1)] = v;
}

// ---------------- fused gather + augmented GEMM via V_WMMA_F32_16X16X4_F32 ----------------
// out[flat, :] = [nodefeat[nodes[flat]] , sums[flat/size]] @ Waug + bias
// block: 256 threads = 8 waves; tile: 48 rows x 128 cols; wave w -> cols [16w,16w+16)
// 3 independent WMMA accumulation chains per wave (rows 0-15 / 16-31 / 32-47)
__global__ __launch_bounds__(256)
void autobahn_gemm_kernel(const float* __restrict__ nodefeat,
                          const int* __restrict__ nodes,
                          const float* __restrict__ sums,
                          const float* __restrict__ wp,
                          const float* __restrict__ bias,
                          float* __restrict__ out,
                          int nsub, int size) {
    __shared__ float Xs[TILE_M * LDX];
    const int M = nsub * size;
    const int tid = threadIdx.x;
    const int blockbase = blockIdx.x * TILE_M;

    // Stage 48 augmented rows (256 floats each) into LDS: 3072 float4 chunks.
    // Gather is per-lane addressed -> use async global->LDS (ASYNCcnt) when available.
    for (int idx = tid; idx < TILE_M * 64; idx += 256) {
        int row = idx >> 6;
        int seg = idx & 63;       // float4 index within 256-float row
        int flat = blockbase + row;
        float* dst = &Xs[row * LDX + seg * 4];
        const float* src = nullptr;
        if (flat < M) {
            if (seg < 32) {
                int node = nodes[flat];
                src = nodefeat + (size_t)node * HID + seg * 4;
            } else {
                int sub = flat / size;
                src = sums + (size_t)sub * HID + (seg - 32) * 4;
            }
        }
#if HAVE_ASYNC_LDS
        if (src) {
            __builtin_amdgcn_global_load_async_to_lds_b128(
                (__attribute__((address_space(1))) v4i*)(uintptr_t)src,
                (__attribute__((address_space(3))) v4i*)(unsigned)(uintptr_t)dst,
                0, 0);
        } else {
            *(float4*)dst = make_float4(0.f, 0.f, 0.f, 0.f);
        }
#else
        float4 v = make_float4(0.f, 0.f, 0.f, 0.f);
        if (src) v = *(const float4*)src;
        *(float4*)dst = v;
#endif
    }
#if HAVE_ASYNC_LDS
#if __has_builtin(__builtin_amdgcn_s_wait_asynccnt)
    __builtin_amdgcn_s_wait_asynccnt(0);
#else
    asm volatile("s_wait_asynccnt 0x0" ::: "memory");
#endif
#endif
    __syncthreads();

    const int lane = tid & 31;
    const int wave = tid >> 5;
    const int lm   = lane & 15;   // A: M index / B,D: N index within tile
    const int hi   = lane >> 4;   // lane half selects K 0/1 vs 2/3 (ISA 7.12.2)
    const int col  = wave * 16 + lm;

    v8f acc[3];
    acc[0] = v8f{}; acc[1] = v8f{}; acc[2] = v8f{};

#pragma unroll 4
    for (int kk = 0; kk < 64; ++kk) {      // K = 256 in steps of 4
        const int k0 = kk * 4;
        const int kA = k0 + hi * 2;
        // B fragment: K striped over lane halves + VGPRs, lane%16 = N; one float2
        v2f b = *(const v2f*)(wp + ((size_t)(kA >> 1) * HID + col) * 2);
        // A fragments: VGPR0 = K0(lanes0-15)/K2(lanes16-31), VGPR1 = K1/K3
        v2f a0 = *(const v2f*)&Xs[(lm +  0) * LDX + kA];
        v2f a1 = *(const v2f*)&Xs[(lm + 16) * LDX + kA];
        v2f a2 = *(const v2f*)&Xs[(lm + 32) * LDX + kA];
        acc[0] = __builtin_amdgcn_wmma_f32_16x16x4_f32(false, a0, false, b,
                                                       (short)0, acc[0], false, false);
        acc[1] = __builtin_amdgcn_wmma_f32_16x16x4_f32(false, a1, false, b,
                                                       (short)0, acc[1], false, false);
        acc[2] = __builtin_amdgcn_wmma_f32_16x16x4_f32(false, a2, false, b,
                                                       (short)0, acc[2], false, false);
    }

    // D layout: VGPR j -> M=j (lanes 0-15) / M=j+8 (lanes 16-31), N = lane%16
    const float bc = bias[col];
    if (blockbase + TILE_M <= M) {
        // full tile: unguarded stores
#pragma unroll
        for (int r = 0; r < 3; ++r) {
            int base = blockbase + 16 * r + (hi ? 8 : 0);
#pragma unroll
            for (int j = 0; j < 8; ++j)
                out[(size_t)(base + j) * HID + col] = acc[r][j] + bc;
        }
    } else {
        // tail tile only
#pragma unroll
        for (int r = 0; r < 3; ++r) {
            int base = blockbase + 16 * r + (hi ? 8 : 0);
#pragma unroll
            for (int j = 0; j < 8; ++j) {
                int f = base + j;
                if (f < M) out[(size_t)f * HID + col] = acc[r][j] + bc;
            }
        }
    }
}

extern "C" void kernel_launch(void* const* d_in, const int* in_sizes, int n_in,
                              void* d_out, int out_size, void* d_ws, size_t ws_size,
                              hipStream_t stream) {
    (void)in_sizes; (void)n_in; (void)out_size; (void)ws_size;
    const int cyc_sizes[2] = {5, 6};
    const int pth_sizes[4] = {4, 5, 6, 7};

    // setup_inputs() dict order: per type: rep, nodes, w_loc, w_glb, b
    const float* c_rep[2]; const int* c_nodes[2];
    const float* c_wl[2];  const float* c_wg[2]; const float* c_b[2];
    for (int i = 0; i < 2; ++i) {
        c_rep[i]   = (const float*)d_in[5 * i + 0];
        c_nodes[i] = (const int*)  d_in[5 * i + 1];
        c_wl[i]    = (const float*)d_in[5 * i + 2];
        c_wg[i]    = (const float*)d_in[5 * i + 3];
        c_b[i]     = (const float*)d_in[5 * i + 4];
    }
    const float* p_rep[4]; const int* p_nodes[4];
    const float* p_wl[4];  const float* p_wg[4]; const float* p_b[4];
    for (int i = 0; i < 4; ++i) {
        p_rep[i]   = (const float*)d_in[10 + 5 * i + 0];
        p_nodes[i] = (const int*)  d_in[10 + 5 * i + 1];
        p_wl[i]    = (const float*)d_in[10 + 5 * i + 2];
        p_wg[i]    = (const float*)d_in[10 + 5 * i + 3];
        p_b[i]     = (const float*)d_in[10 + 5 * i + 4];
    }

    // workspace layout (floats)
    float* ws = (float*)d_ws;
    size_t off = 0;
    float* f_cyc = ws + off; off += (size_t)NN * HID;   // cycle-pooled node feats
    float* f_pth = ws + off; off += (size_t)NN * HID;   // path-pooled node feats
    float* sum_c[2]; for (int i = 0; i < 2; ++i) { sum_c[i] = ws + off; off += (size_t)NCYC  * HID; }
    float* sum_p[4]; for (int i = 0; i < 4; ++i) { sum_p[i] = ws + off; off += (size_t)NPATH * HID; }
    float* wp_c[2];  for (int i = 0; i < 2; ++i) { wp_c[i]  = ws + off; off += (size_t)2 * HID * HID; }
    float* wp_p[4];  for (int i = 0; i < 4; ++i) { wp_p[i]  = ws + off; off += (size_t)2 * HID * HID; }

    // 1) zero both node tables (contiguous) — every call, no stale state
    (void)hipMemsetAsync(f_cyc, 0, (size_t)2 * NN * HID * sizeof(float), stream);

    // 2) scatter-add pools
    for (int i = 0; i < 2; ++i) {
        int nrows = NCYC * cyc_sizes[i];
        scatter_add_kernel<<<nrows, HID, 0, stream>>>(c_rep[i], c_nodes[i], f_cyc, nrows);
    }
    for (int i = 0; i < 4; ++i) {
        int nrows = NPATH * pth_sizes[i];
        scatter_add_kernel<<<nrows, HID, 0, stream>>>(p_rep[i], p_nodes[i], f_pth, nrows);
    }

    // 3) per-subgraph sums (cycles gather path-pool; paths gather cycle-pool)
    for (int i = 0; i < 2; ++i)
        subgraph_sum_kernel<<<NCYC, HID, 0, stream>>>(f_pth, c_nodes[i], sum_c[i], NCYC, cyc_sizes[i]);
    for (int i = 0; i < 4; ++i)
        subgraph_sum_kernel<<<NPATH, HID, 0, stream>>>(f_cyc, p_nodes[i], sum_p[i], NPATH, pth_sizes[i]);

    // 4) effective packed weights
    const int wblk = (2 * HID * HID) / 256;
    for (int i = 0; i < 2; ++i)
        pack_weights_kernel<<<wblk, 256, 0, stream>>>(c_wl[i], c_wg[i], wp_c[i], (float)cyc_sizes[i]);
    for (int i = 0; i < 4; ++i)
        pack_weights_kernel<<<wblk, 256, 0, stream>>>(p_wl[i], p_wg[i], wp_p[i], (float)pth_sizes[i]);

    // 5) fused gather + WMMA GEMM, outputs concatenated in return order
    float* out = (float*)d_out;
    size_t o = 0;
    for (int i = 0; i < 2; ++i) {
        int Mrows = NCYC * cyc_sizes[i];
        int blocks = (Mrows + TILE_M - 1) / TILE_M;
        autobahn_gemm_kernel<<<blocks, 256, 0, stream>>>(f_pth, c_nodes[i], sum_c[i],
                                                         wp_c[i], c_b[i], out + o,
                                                         NCYC, cyc_sizes[i]);
        o += (size_t)Mrows * HID;
    }
    for (int i = 0; i < 4; ++i) {
        int Mrows = NPATH * pth_sizes[i];
        int blocks = (Mrows + TILE_M - 1) / TILE_M;
        autobahn_gemm_kernel<<<blocks, 256, 0, stream>>>(f_cyc, p_nodes[i], sum_p[i],
                                                         wp_p[i], p_b[i], out + o,
                                                         NPATH, pth_sizes[i]);
        o += (size_t)Mrows * HID;
    }
}